// NGRB_55216099558072
// MI455X (gfx1250) — compile-verified
//
#include <hip/hip_runtime.h>
#include <hip/hip_bf16.h>

typedef float v4f __attribute__((ext_vector_type(4)));
typedef int   gv4i __attribute__((vector_size(16)));   // matches async builtin's pointee type

#define AS1 __attribute__((address_space(1)))
#define AS3 __attribute__((address_space(3)))

#define SEQ_CAP    1024   // max staged history tokens (problem uses 512)
#define MASK_WORDS 2048   // supports vocab up to 65536 bits (problem: 50257)

#if __has_builtin(__builtin_amdgcn_global_load_async_to_lds_b128)
#define HAVE_ASYNC_LDS 1
#else
#define HAVE_ASYNC_LDS 0
#endif

__device__ __forceinline__ void wait_asynccnt0() {
#if __has_builtin(__builtin_amdgcn_s_wait_asynccnt)
  __builtin_amdgcn_s_wait_asynccnt(0);
#else
  asm volatile("s_wait_asynccnt 0" ::: "memory");
#endif
}

__global__ __launch_bounds__(512) void ngram_ban_copy_kernel(
    const int* __restrict__ tokens, const float* __restrict__ lprobs,
    float* __restrict__ out,
    const int* __restrict__ p_bsz, const int* __restrict__ p_step,
    const int* __restrict__ p_beam, const int* __restrict__ p_n,
    int tokens_flat, int lprobs_flat)
{
  __shared__ int      s_tok[SEQ_CAP];
  __shared__ unsigned s_mask[MASK_WORDS];

  const int bsz  = p_bsz[0];
  const int step = p_step[0];
  const int beam = p_beam[0];
  const int n    = p_n[0];
  const int B    = bsz * beam;
  // 32-bit unsigned division: flat counts come from int in_sizes[] anyway.
  const int seq_len = (int)((unsigned)tokens_flat / (unsigned)B);
  const int vocab   = (int)((unsigned)lprobs_flat / (unsigned)B);

  int hist_len = step + 1;
  if (hist_len > seq_len) hist_len = seq_len;
  if (hist_len > SEQ_CAP) hist_len = SEQ_CAP;
  int mwords = (vocab + 31) >> 5;
  if (mwords > MASK_WORDS) mwords = MASK_WORDS;

  const int W    = step + 2 - n;   // number of candidate windows (may be <= 0)
  const int tid  = threadIdx.x;
  const int nthr = blockDim.x;
  const int nvec = vocab >> 2;
  const float NEG_INF = -__builtin_inff();

  for (int row = blockIdx.x; row < B; row += gridDim.x) {
    const int*   trow = tokens + (size_t)row * (size_t)seq_len;
    const v4f*   src4 = (const v4f*)(lprobs + (size_t)row * (size_t)vocab);
    v4f*         dst4 = (v4f*)(out + (size_t)row * (size_t)vocab);
    const float* src  = lprobs + (size_t)row * (size_t)vocab;
    float*       dst  = out    + (size_t)row * (size_t)vocab;

    // ---- stage token history into LDS via CDNA5 async global->LDS DMA ----
#if HAVE_ASYNC_LDS
    for (int t = tid; t < (hist_len >> 2); t += nthr) {
      __builtin_amdgcn_global_load_async_to_lds_b128(
          (AS1 gv4i*)(trow + 4 * t),
          (AS3 gv4i*)&s_tok[4 * t],
          /*offset=*/0, /*cpol=*/0);
    }
    for (int t = (hist_len & ~3) + tid; t < hist_len; t += nthr)
      s_tok[t] = trow[t];
#else
    for (int t = tid; t < hist_len; t += nthr) s_tok[t] = trow[t];
#endif

    // zero the banned-token bitmask while the token DMA is in flight
    for (int m = tid; m < mwords; m += nthr) s_mask[m] = 0u;

#if HAVE_ASYNC_LDS
    wait_asynccnt0();
#endif
    __syncthreads();

    // ---- scan windows vs. the (n-1)-token prefix; mark banned tokens ----
    if (W > 0 && n >= 1) {
      for (int w = tid; w < W; w += nthr) {
        bool match = true;
        for (int j = 0; j < n - 1; ++j)
          match = match && (s_tok[w + j] == s_tok[W + j]);
        if (match) {
          int bt = s_tok[w + n - 1];
          if ((unsigned)bt < (unsigned)vocab && (bt >> 5) < mwords)
            atomicOr(&s_mask[bt >> 5], 1u << (bt & 31));
        }
      }
    }
    __syncthreads();

    // ---- fused streaming copy + ban apply; 128-bit NT accesses ----
#pragma unroll 4
    for (int i = tid; i < nvec; i += nthr) {
      v4f v = __builtin_nontemporal_load(&src4[i]);
      unsigned bits = (s_mask[i >> 3] >> ((i & 7) * 4)) & 0xFu;
      if (bits) {
        if (bits & 1u) v.x = NEG_INF;
        if (bits & 2u) v.y = NEG_INF;
        if (bits & 4u) v.z = NEG_INF;
        if (bits & 8u) v.w = NEG_INF;
      }
      __builtin_nontemporal_store(v, &dst4[i]);
    }
    for (int i = (nvec << 2) + tid; i < vocab; i += nthr) {
      float v = src[i];
      if ((s_mask[i >> 5] >> (i & 31)) & 1u) v = NEG_INF;
      __builtin_nontemporal_store(v, &dst[i]);
    }
    __syncthreads();   // protect s_tok/s_mask reuse on next row
  }
}

extern "C" void kernel_launch(void* const* d_in, const int* in_sizes, int n_in,
                              void* d_out, int out_size, void* d_ws, size_t ws_size,
                              hipStream_t stream) {
  (void)n_in; (void)d_ws; (void)ws_size; (void)out_size;
  const int*   tokens = (const int*)d_in[0];
  const float* lprobs = (const float*)d_in[1];
  const int*   p_bsz  = (const int*)d_in[2];
  const int*   p_step = (const int*)d_in[3];
  const int*   p_beam = (const int*)d_in[4];
  const int*   p_n    = (const int*)d_in[5];
  float*       out    = (float*)d_out;

  // 512 blocks x 512 threads (16 wave32s/block); row-strided loop covers any B.
  dim3 grid(512), block(512);
  ngram_ban_copy_kernel<<<grid, block, 0, stream>>>(
      tokens, lprobs, out, p_bsz, p_step, p_beam, p_n,
      in_sizes[0], in_sizes[1]);
}